// HierarchicalPrototypeBank_38800734552297
// MI455X (gfx1250) — compile-verified
//
#include <hip/hip_runtime.h>
#include <math.h>

typedef float v2f __attribute__((ext_vector_type(2)));
typedef float v4f __attribute__((ext_vector_type(4)));
typedef float v8f __attribute__((ext_vector_type(8)));

__device__ __forceinline__ float gelu_exact(float x) {
    return 0.5f * x * (1.0f + erff(x * 0.7071067811865476f));
}

// ---------------------------------------------------------------------------
// 1) pooled[b*768+c] = mean over 64*64 spatial elements (bandwidth-bound).
//    403 MB read-once stream > 192 MB L2 -> non-temporal loads so the hot
//    small tensors (banks, w_query) stay resident.
// ---------------------------------------------------------------------------
__global__ __launch_bounds__(256) void pool_kernel(const float* __restrict__ feat,
                                                   float* __restrict__ pooled) {
    __shared__ float red[256];
    const int row = blockIdx.x;  // b*768 + c, 0..24575
    const v4f* in4 = reinterpret_cast<const v4f*>(feat + (size_t)row * 4096);
    float acc = 0.f;
    for (int i = threadIdx.x; i < 1024; i += 256) {
        v4f q = __builtin_nontemporal_load(&in4[i]);
        acc += q.x + q.y + q.z + q.w;
    }
    red[threadIdx.x] = acc;
    __syncthreads();
    for (int s = 128; s > 0; s >>= 1) {
        if (threadIdx.x < s) red[threadIdx.x] += red[threadIdx.x + s];
        __syncthreads();
    }
    if (threadIdx.x == 0) pooled[row] = red[0] * (1.0f / 4096.0f);
}

// ---------------------------------------------------------------------------
// 2) token_raw = pooled @ Wq^T  via V_WMMA_F32_16X16X4_F32 (one wave / tile)
//    M=32 (2 tiles), N=768 (48 tiles), K=768 (192 steps of 4)
// ---------------------------------------------------------------------------
__global__ __launch_bounds__(32) void token_gemm(const float* __restrict__ pooled,
                                                 const float* __restrict__ wq,
                                                 float* __restrict__ traw) {
    const int tile = blockIdx.x;           // 0..95
    const int m0 = (tile / 48) * 16;
    const int n0 = (tile % 48) * 16;
    const int lane = threadIdx.x;
    const int hf = lane >> 4;              // lane half (0/1)
    const int lr = lane & 15;
    const float* arow = pooled + (m0 + lr) * 768;  // A: M = lane%16
    const float* brow = wq + (n0 + lr) * 768;      // B[K=c][N=d] = Wq[d*768+c]
    v8f acc = {};
    for (int k0 = 0; k0 < 768; k0 += 4) {
        const int ka = k0 + 2 * hf;        // A/B frag: VGPR v holds K = k0 + v + 2*half
        v2f a; a.x = arow[ka]; a.y = arow[ka + 1];
        v2f b; b.x = brow[ka]; b.y = brow[ka + 1];
        acc = __builtin_amdgcn_wmma_f32_16x16x4_f32(false, a, false, b,
                                                    (short)0, acc, false, false);
    }
    // D layout: VGPR v, lanes0-15 -> M=v, lanes16-31 -> M=v+8; N = lane%16
    for (int v = 0; v < 8; ++v)
        traw[(m0 + v + 8 * hf) * 768 + n0 + lr] = acc[v];
}

// ---------------------------------------------------------------------------
// 3) routing MLP + token normalize (single block)
// ---------------------------------------------------------------------------
__global__ __launch_bounds__(256) void head_kernel(
    const float* __restrict__ pooled, const float* __restrict__ traw,
    const float* __restrict__ rw1, const float* __restrict__ rb1,
    const float* __restrict__ rw2, const float* __restrict__ rb2,
    float* __restrict__ out_lw, float* __restrict__ out_token) {
    __shared__ float s_h[32 * 128];
    __shared__ float s_scale[32];
    const int tid = threadIdx.x;
    for (int idx = tid; idx < 32 * 128; idx += 256) {
        const int b = idx >> 7, j = idx & 127;
        float acc = rb1[j];
        const float* pr = pooled + b * 768;
        for (int c = 0; c < 768; ++c) acc += pr[c] * rw1[c * 128 + j];
        s_h[idx] = gelu_exact(acc);
    }
    __syncthreads();
    if (tid < 32) {
        const int b = tid;
        float lg[3];
        for (int l = 0; l < 3; ++l) {
            float a = rb2[l];
            for (int j = 0; j < 128; ++j) a += s_h[b * 128 + j] * rw2[j * 3 + l];
            lg[l] = a;
        }
        const float m = fmaxf(lg[0], fmaxf(lg[1], lg[2]));
        const float e0 = expf(lg[0] - m), e1 = expf(lg[1] - m), e2 = expf(lg[2] - m);
        const float inv = 1.0f / (e0 + e1 + e2);
        out_lw[b * 3 + 0] = e0 * inv;
        out_lw[b * 3 + 1] = e1 * inv;
        out_lw[b * 3 + 2] = e2 * inv;
        float ss = 0.f;
        const float* tr = traw + b * 768;
        for (int d = 0; d < 768; ++d) ss += tr[d] * tr[d];
        s_scale[b] = 1.0f / fmaxf(sqrtf(ss), 1e-12f);
    }
    __syncthreads();
    for (int idx = tid; idx < 32 * 768; idx += 256)
        out_token[idx] = traw[idx] * s_scale[idx / 768];
}

// ---------------------------------------------------------------------------
// 4) g[l] = token @ mg_w1[l]  (WMMA f32; M=32, N=384, K=768, per level)
// ---------------------------------------------------------------------------
__global__ __launch_bounds__(32) void gate_gemm(const float* __restrict__ token,
                                                const float* __restrict__ mg_w1,
                                                float* __restrict__ g) {
    const int l = blockIdx.x / 48;
    const int tile = blockIdx.x % 48;
    const int m0 = (tile / 24) * 16;
    const int n0 = (tile % 24) * 16;
    const int lane = threadIdx.x;
    const int hf = lane >> 4;
    const int lr = lane & 15;
    const float* w1 = mg_w1 + l * 768 * 384;   // [K=d][N=h] row-major, stride 384
    const float* arow = token + (m0 + lr) * 768;
    v8f acc = {};
    for (int k0 = 0; k0 < 768; k0 += 4) {
        const int ka = k0 + 2 * hf;
        v2f a; a.x = arow[ka]; a.y = arow[ka + 1];
        v2f b;
        b.x = w1[ka * 384 + n0 + lr];
        b.y = w1[(ka + 1) * 384 + n0 + lr];
        acc = __builtin_amdgcn_wmma_f32_16x16x4_f32(false, a, false, b,
                                                    (short)0, acc, false, false);
    }
    float* gout = g + l * 32 * 384;
    for (int v = 0; v < 8; ++v)
        gout[(m0 + v + 8 * hf) * 384 + n0 + lr] = acc[v];
}

// ---------------------------------------------------------------------------
// 5) per (bank,level): mix gate, bank norms, logits, softmax, entropy
//    contribution (private slot -> deterministic), weighted attention scratch
// ---------------------------------------------------------------------------
__global__ __launch_bounds__(256) void bank_kernel(
    const float* __restrict__ token, const float* __restrict__ g,
    const float* __restrict__ mg_b1, const float* __restrict__ mg_w2,
    const float* __restrict__ mg_b2,
    const float* __restrict__ fast_banks, const float* __restrict__ slow_banks,
    const float* __restrict__ fast_counts, const float* __restrict__ slow_counts,
    const float* __restrict__ lw,
    float* __restrict__ out_mix, float* __restrict__ out_attn,
    float* __restrict__ attnw, float* __restrict__ entp) {
    const int combo = blockIdx.x;                 // 0..5
    const int is_fast = (combo < 3) ? 1 : 0;
    const int l = combo % 3;
    const float* bank = (is_fast ? fast_banks : slow_banks) + l * 64 * 768;
    const float* counts = (is_fast ? fast_counts : slow_counts) + l * 64;
    __shared__ float s_red[256];
    __shared__ float s_mix[32];
    __shared__ float s_inv[64];
    __shared__ float s_lg[32 * 64];
    const int tid = threadIdx.x;

    {   // mix-gate: gelu(g + b1) . w2, 8 partials per batch row
        const int b = tid >> 3, part = tid & 7;
        float acc = 0.f;
        const float* gb = g + (l * 32 + b) * 384;
        const float* b1 = mg_b1 + l * 384;
        const float* w2 = mg_w2 + l * 384;
        for (int hh = part * 48; hh < part * 48 + 48; ++hh)
            acc += gelu_exact(gb[hh] + b1[hh]) * w2[hh];
        s_red[tid] = acc;
    }
    if (tid < 64) {   // bank row inverse norms
        const float* br = bank + tid * 768;
        float ss = 0.f;
        for (int d = 0; d < 768; ++d) ss += br[d] * br[d];
        s_inv[tid] = 1.0f / fmaxf(sqrtf(ss), 1e-12f);
    }
    __syncthreads();
    if (tid < 32) {
        float a = 0.f;
        for (int p = 0; p < 8; ++p) a += s_red[tid * 8 + p];
        const float mix = 1.0f / (1.0f + expf(-(a + mg_b2[l])));
        s_mix[tid] = mix;
        if (is_fast) {
            out_mix[tid * 6 + l * 2 + 0] = mix;
            out_mix[tid * 6 + l * 2 + 1] = 1.0f - mix;
        }
    }
    __syncthreads();
    const float invSqrtD = 0.036084391824351615f;  // 1/sqrt(768)
    for (int o = tid; o < 2048; o += 256) {        // logits [32,64]
        const int b = o >> 6, k = o & 63;
        const float* tr = token + b * 768;
        const float* br = bank + k * 768;
        float acc = 0.f;
        for (int d = 0; d < 768; ++d) acc += tr[d] * br[d];
        s_lg[o] = (counts[k] > 1e-6f) ? acc * s_inv[k] * invSqrtD : -INFINITY;
    }
    __syncthreads();
    if (tid < 32) {   // softmax + entropy + weighted attn
        const int b = tid;
        float* row = s_lg + b * 64;
        float m = -INFINITY;
        for (int k = 0; k < 64; ++k) m = fmaxf(m, row[k]);
        float denom = 0.f;
        for (int k = 0; k < 64; ++k) { float e = expf(row[k] - m); row[k] = e; denom += e; }
        const float inv = 1.0f / denom;
        const float lwv = lw[b * 3 + l];
        const float wctx = lwv * (is_fast ? s_mix[b] : (1.0f - s_mix[b]));
        float ent = 0.f;
        float* ao = out_attn + ((l * 32 + b) * 2 + (is_fast ? 0 : 1)) * 64;
        float* aw = attnw + combo * 2048 + b * 64;
        for (int k = 0; k < 64; ++k) {
            const float attn = row[k] * inv;
            ao[k] = attn;
            aw[k] = wctx * attn * s_inv[k];
            if (counts[k] > 1e-6f) {
                const float p = fmaxf(attn, 1e-8f);
                ent -= p * logf(p);
            }
        }
        entp[combo * 32 + b] = lwv * 0.5f * ent;   // private slot, no atomics
    }
}

// ---------------------------------------------------------------------------
// 6) context_global[b,d] = sum_{6 banks} sum_k aw[b,k] * bank[k,d];
//    entropy = fixed-order sum of the 6 per-combo contributions
// ---------------------------------------------------------------------------
__global__ __launch_bounds__(256) void context_kernel(
    const float* __restrict__ attnw,
    const float* __restrict__ fast_banks, const float* __restrict__ slow_banks,
    const float* __restrict__ entp,
    float* __restrict__ out_ctxg, float* __restrict__ out_ent) {
    const int gid = blockIdx.x * 256 + threadIdx.x;  // 0..24575
    const int b = gid / 768, d = gid % 768;
    float acc = 0.f;
    for (int combo = 0; combo < 6; ++combo) {
        const float* bank = ((combo < 3) ? fast_banks : slow_banks) + (combo % 3) * 64 * 768;
        const float* aw = attnw + combo * 2048 + b * 64;
        for (int k = 0; k < 64; ++k) acc += aw[k] * bank[k * 768 + d];
    }
    out_ctxg[gid] = acc;
    if (blockIdx.x == 0 && threadIdx.x < 32) {
        float e = 0.f;
        for (int c = 0; c < 6; ++c) e += entp[c * 32 + threadIdx.x];
        out_ent[threadIdx.x] = e;
    }
}

// ---------------------------------------------------------------------------
// 7) context_spatial broadcast: 403 MB write-once stream -> non-temporal
//    stores keep L2 clean for subsequent replays.
// ---------------------------------------------------------------------------
__global__ __launch_bounds__(256) void broadcast_kernel(const float* __restrict__ ctxg,
                                                        float* __restrict__ out) {
    const int row = blockIdx.x;  // b*768 + c
    const float v = ctxg[row];
    v4f v4; v4.x = v; v4.y = v; v4.z = v; v4.w = v;
    v4f* o4 = reinterpret_cast<v4f*>(out + (size_t)row * 4096);
    for (int i = threadIdx.x; i < 1024; i += 256)
        __builtin_nontemporal_store(v4, &o4[i]);
}

extern "C" void kernel_launch(void* const* d_in, const int* in_sizes, int n_in,
                              void* d_out, int out_size, void* d_ws, size_t ws_size,
                              hipStream_t stream) {
    const float* features    = (const float*)d_in[0];
    const float* w_query     = (const float*)d_in[1];
    const float* rw1         = (const float*)d_in[2];
    const float* rb1         = (const float*)d_in[3];
    const float* rw2         = (const float*)d_in[4];
    const float* rb2         = (const float*)d_in[5];
    const float* mg_w1       = (const float*)d_in[6];
    const float* mg_b1       = (const float*)d_in[7];
    const float* mg_w2       = (const float*)d_in[8];
    const float* mg_b2       = (const float*)d_in[9];
    const float* fast_banks  = (const float*)d_in[10];
    const float* slow_banks  = (const float*)d_in[11];
    const float* fast_counts = (const float*)d_in[12];
    const float* slow_counts = (const float*)d_in[13];

    float* out       = (float*)d_out;
    float* o_spatial = out;                  // [32,768,64,64]
    float* o_global  = out + 100663296;      // [32,768]
    float* o_lw      = out + 100687872;      // [32,3]
    float* o_mix     = out + 100687968;      // [32,3,2]
    float* o_ent     = out + 100688160;      // [32]
    float* o_token   = out + 100688192;      // [32,768]
    float* o_attn    = out + 100712768;      // [3,32,2,64]

    float* ws       = (float*)d_ws;
    float* w_pooled = ws;                    // 24576 floats
    float* w_traw   = ws + 24576;            // 24576
    float* w_g      = ws + 49152;            // 36864
    float* w_attnw  = ws + 86016;            // 12288
    float* w_entp   = ws + 98304;            // 6*32 = 192

    pool_kernel<<<24576, 256, 0, stream>>>(features, w_pooled);
    token_gemm<<<96, 32, 0, stream>>>(w_pooled, w_query, w_traw);
    head_kernel<<<1, 256, 0, stream>>>(w_pooled, w_traw, rw1, rb1, rw2, rb2,
                                       o_lw, o_token);
    gate_gemm<<<144, 32, 0, stream>>>(o_token, mg_w1, w_g);
    bank_kernel<<<6, 256, 0, stream>>>(o_token, w_g, mg_b1, mg_w2, mg_b2,
                                       fast_banks, slow_banks, fast_counts, slow_counts,
                                       o_lw, o_mix, o_attn, w_attnw, w_entp);
    context_kernel<<<96, 256, 0, stream>>>(w_attnw, fast_banks, slow_banks, w_entp,
                                           o_global, o_ent);
    broadcast_kernel<<<24576, 256, 0, stream>>>(o_global, o_spatial);
}